// HierarchicalNetwork_42992622633709
// MI455X (gfx1250) — compile-verified
//
#include <hip/hip_runtime.h>
#include <hip/hip_bf16.h>

#define PRE   1024
#define POST  1024
#define BATCH 256
#define HIST  50

typedef __attribute__((ext_vector_type(2))) float v2f;
typedef __attribute__((ext_vector_type(8))) float v8f;

// ---------------------------------------------------------------------------
// Kernel 1: synaptic_current = pre_spikes @ (W * mask), f32 WMMA 16x16x4.
// One wave32 per 16x16 output tile. M=256 -> 16 tiles, N=1024 -> 64 tiles.
// 8 waves / block, 128 blocks.
// ---------------------------------------------------------------------------
__global__ __launch_bounds__(256) void snn_gemm_wmma_f32(
    const float* __restrict__ A,    // pre_spikes [BATCH][PRE]
    const float* __restrict__ W,    // synaptic_weights [PRE][POST]
    const float* __restrict__ Mk,   // connectivity_mask [PRE][POST]
    float* __restrict__ C)          // synaptic_current [BATCH][POST]
{
    const int lane = threadIdx.x & 31;
    const int wave = threadIdx.x >> 5;                 // 0..7
    const int tile = blockIdx.x * 8 + wave;            // 0..1023
    const int tileM = tile >> 6;                       // 0..15   (POST/16 = 64 tiles in N)
    const int tileN = tile & 63;                       // 0..63
    const int m0 = tileM * 16;
    const int n0 = tileN * 16;

    const int half = lane >> 4;                        // 0: lanes 0-15, 1: lanes 16-31
    const int lid  = lane & 15;

    v8f acc = {};                                      // 16x16 f32 accumulator (8 VGPRs)

    // A fragment row for this lane: M = m0 + lid (both halves cover same M range)
    const float* arow = A + (size_t)(m0 + lid) * PRE;

    for (int k = 0; k < PRE; k += 4) {
        // A 16x4 (MxK): v0/v1 = K {k,k+1} for lanes 0-15, {k+2,k+3} for lanes 16-31.
        // K is contiguous in the spike row -> single 8B load.
        const float2 av = *(const float2*)(arow + k + half * 2);
        v2f afrag; afrag.x = av.x; afrag.y = av.y;

        // B 4x16 (KxN): lanes 0-15 hold rows K={k,k+1}, lanes 16-31 rows K={k+2,k+3}.
        // Fuse the connectivity mask into the operand (effective_weights).
        const int kr = k + half * 2;
        const size_t b0 = (size_t)kr * POST + n0 + lid;
        v2f bfrag;
        bfrag.x = W[b0]        * Mk[b0];
        bfrag.y = W[b0 + POST] * Mk[b0 + POST];

        // D = A*B + C  (8 args: neg_a, A, neg_b, B, c_mod, C, reuse_a, reuse_b)
        acc = __builtin_amdgcn_wmma_f32_16x16x4_f32(
            false, afrag, false, bfrag, (short)0, acc, false, false);
    }

    // C/D layout: VGPR r -> M = m0 + half*8 + r, N = n0 + lid
#pragma unroll
    for (int r = 0; r < 8; ++r) {
        C[(size_t)(m0 + half * 8 + r) * POST + n0 + lid] = acc[r];
    }
}

// ---------------------------------------------------------------------------
// Kernel 2: column means over batch + EMA buffers.
// j in [0,1024): pre side; j in [1024,2048): post side.
// Consecutive threads read consecutive columns -> fully coalesced.
// ---------------------------------------------------------------------------
__global__ __launch_bounds__(256) void snn_activity(
    const float* __restrict__ pre_spikes,   // [BATCH][PRE]
    const float* __restrict__ post_spikes,  // [BATCH][POST]
    const float* __restrict__ pre_act,      // [PRE]
    const float* __restrict__ post_act,     // [POST]
    float* __restrict__ new_pre_act,        // [PRE]  (output)
    float* __restrict__ new_post_act,       // [POST] (output)
    float* __restrict__ pre_mean,           // [PRE]  (workspace)
    float* __restrict__ post_mean)          // [POST] (workspace)
{
    const int j = blockIdx.x * blockDim.x + threadIdx.x;
    if (j < PRE) {
        float s = 0.0f;
        for (int b = 0; b < BATCH; ++b) s += pre_spikes[(size_t)b * PRE + j];
        const float m = s * (1.0f / BATCH);
        pre_mean[j]    = m;
        new_pre_act[j] = 0.99f * pre_act[j] + 0.01f * m;
    } else {
        const int jj = j - PRE;
        float s = 0.0f;
        for (int b = 0; b < BATCH; ++b) s += post_spikes[(size_t)b * POST + jj];
        const float m = s * (1.0f / BATCH);
        post_mean[jj]    = m;
        new_post_act[jj] = 0.99f * post_act[jj] + 0.01f * m;
    }
}

// ---------------------------------------------------------------------------
// Kernel 3: fused circular-buffer scatter + copy + mean.
// One thread per (i,j) pair: stream 50 contiguous f32 once, substitute the
// slot `corr_index % 50` with pre_mean[i]*post_mean[j], emit new_history and
// average_correlation in the same pass (history read exactly once).
// ---------------------------------------------------------------------------
__global__ __launch_bounds__(256) void snn_history(
    const float* __restrict__ hist,        // [PRE][POST][HIST]
    const float* __restrict__ pre_mean,    // [PRE]
    const float* __restrict__ post_mean,   // [POST]
    const int*   __restrict__ corr_index,  // scalar
    float* __restrict__ new_hist,          // [PRE][POST][HIST] (output)
    float* __restrict__ avg)               // [PRE][POST]       (output)
{
    const int pair = blockIdx.x * blockDim.x + threadIdx.x;   // 0 .. PRE*POST-1
    const int i = pair >> 10;
    const int j = pair & (POST - 1);
    const int idx = ((corr_index[0] % HIST) + HIST) % HIST;

    const float cur = pre_mean[i] * post_mean[j];

    const float* h  = hist     + (size_t)pair * HIST;
    float*       nh = new_hist + (size_t)pair * HIST;

    float sum = 0.0f;
#pragma unroll
    for (int t = 0; t < HIST; ++t) {
        const float v = (t == idx) ? cur : h[t];
        nh[t] = v;
        sum += v;
    }
    avg[pair] = sum * (1.0f / HIST);
}

// ---------------------------------------------------------------------------
extern "C" void kernel_launch(void* const* d_in, const int* in_sizes, int n_in,
                              void* d_out, int out_size, void* d_ws, size_t ws_size,
                              hipStream_t stream) {
    const float* pre_spikes  = (const float*)d_in[0];  // [256,1024]
    const float* post_spikes = (const float*)d_in[1];  // [256,1024]
    const float* weights     = (const float*)d_in[2];  // [1024,1024]
    const float* mask        = (const float*)d_in[3];  // [1024,1024]
    const float* hist        = (const float*)d_in[4];  // [1024,1024,50]
    const float* pre_act     = (const float*)d_in[5];  // [1024]
    const float* post_act    = (const float*)d_in[6];  // [1024]
    const int*   corr_index  = (const int*)d_in[7];    // scalar

    // Outputs concatenated flat in return order:
    float* out          = (float*)d_out;
    float* syn_current  = out;                                     // 256*1024
    float* new_pre_act  = syn_current + (size_t)BATCH * POST;      // 1024
    float* new_post_act = new_pre_act + PRE;                       // 1024
    float* avg_corr     = new_post_act + POST;                     // 1024*1024
    float* new_history  = avg_corr + (size_t)PRE * POST;           // 1024*1024*50

    float* pre_mean  = (float*)d_ws;        // 1024 f32
    float* post_mean = pre_mean + PRE;      // 1024 f32

    // GEMM: 1024 tiles of 16x16, 8 waves/block -> 128 blocks
    snn_gemm_wmma_f32<<<128, 256, 0, stream>>>(pre_spikes, weights, mask, syn_current);

    // Means + EMA: 2048 threads
    snn_activity<<<8, 256, 0, stream>>>(pre_spikes, post_spikes, pre_act, post_act,
                                        new_pre_act, new_post_act, pre_mean, post_mean);

    // History scatter + copy + mean: 1M threads
    snn_history<<<(PRE * POST) / 256, 256, 0, stream>>>(hist, pre_mean, post_mean,
                                                        corr_index, new_history, avg_corr);
}